// AdditiveAttention_69767448756675
// MI455X (gfx1250) — compile-verified
//
#include <hip/hip_runtime.h>
#include <hip/hip_bf16.h>

typedef __attribute__((ext_vector_type(2))) float v2f;
typedef __attribute__((ext_vector_type(4))) float v4f;
typedef __attribute__((ext_vector_type(8))) float v8f;

#define B_  4
#define Q_  128
#define KV_ 1024
#define D_  256
#define H_  256

__device__ __forceinline__ float fast_tanh(float x) {
#if __has_builtin(__builtin_amdgcn_tanhf)
  return __builtin_amdgcn_tanhf(x);   // V_TANH_F32 (gfx1250 trans op)
#else
  return tanhf(x);
#endif
}

// ---------------------------------------------------------------------------
// P[R x 256] = X[R x 256] * W[256 x 256], fp32, via V_WMMA_F32_16X16X4_F32.
// One wave per 16(M) x 64(N) output tile; 4 accumulators share the A fragment.
// Fragment layouts per CDNA5 ISA 7.12.2:
//   A (16x4 f32):  lane L -> M = L&15,  a[v] = A[M][k + 2*(L>>4) + v]
//   B (4x16 f32):  lane L -> N = L&15,  b[v] = B[k + 2*(L>>4) + v][N]
//   C/D (16x16):   lane L, vgpr v -> M = v + 8*(L>>4), N = L&15
// ---------------------------------------------------------------------------
__global__ __launch_bounds__(32)
void proj_gemm_wmma(const float* __restrict__ X, const float* __restrict__ W,
                    float* __restrict__ P) {
  const int lane = threadIdx.x & 31;
  const int m    = lane & 15;
  const int half = lane >> 4;
  const int row0 = blockIdx.x * 16;
  const int col0 = blockIdx.y * 64;

  v8f acc0 = {}, acc1 = {}, acc2 = {}, acc3 = {};
  const float* xrow = X + (size_t)(row0 + m) * H_;

  for (int k = 0; k < H_; k += 4) {
    const int ka = k + 2 * half;
    v2f a = *(const v2f*)(xrow + ka);                  // global_load_b64
    const float* w0 = W + (size_t)ka * H_ + col0 + m;  // row ka
    const float* w1 = w0 + H_;                         // row ka+1
    v2f b0, b1, b2, b3;
    b0.x = w0[0];  b0.y = w1[0];
    b1.x = w0[16]; b1.y = w1[16];
    b2.x = w0[32]; b2.y = w1[32];
    b3.x = w0[48]; b3.y = w1[48];
    acc0 = __builtin_amdgcn_wmma_f32_16x16x4_f32(false, a, false, b0, (short)0, acc0, false, false);
    acc1 = __builtin_amdgcn_wmma_f32_16x16x4_f32(false, a, false, b1, (short)0, acc1, false, false);
    acc2 = __builtin_amdgcn_wmma_f32_16x16x4_f32(false, a, false, b2, (short)0, acc2, false, false);
    acc3 = __builtin_amdgcn_wmma_f32_16x16x4_f32(false, a, false, b3, (short)0, acc3, false, false);
  }

  float* prow = P + (size_t)(row0 + 8 * half) * H_ + col0 + m;
#pragma unroll
  for (int v = 0; v < 8; ++v) {
    prow[(size_t)v * H_ +  0] = acc0[v];
    prow[(size_t)v * H_ + 16] = acc1[v];
    prow[(size_t)v * H_ + 32] = acc2[v];
    prow[(size_t)v * H_ + 48] = acc3[v];
  }
}

// ---------------------------------------------------------------------------
// Fused additive-attention scores + masked online softmax + attn @ V.
// One block (4 waves) per (b,q). Wave w handles kv = w, w+4, ... < valid_len.
// Masked kv (>= valid_len) would get score -1e6 -> exp underflows to exactly
// 0 in fp32 (real scores are O(10)), so skipping them is bit-equivalent.
// Each lane owns 8 contiguous h/v channels (hbase = lane*8).
// ---------------------------------------------------------------------------
__global__ __launch_bounds__(128)
void attn_fused(const float* __restrict__ qp, const float* __restrict__ kp,
                const float* __restrict__ values, const int* __restrict__ valid_lens,
                const float* __restrict__ w_v, float* __restrict__ out) {
  const int bq    = blockIdx.x;          // 0..511
  const int b     = bq >> 7;             // / Q_
  const int wid   = threadIdx.x >> 5;    // wave in block: 0..3
  const int lane  = threadIdx.x & 31;
  const int hbase = lane * 8;

  __shared__ float sm_m[4];
  __shared__ float sm_l[4];
  __shared__ float sm_acc[4][D_];

  const float* qrow = qp + (size_t)bq * H_ + hbase;
  v4f qv0 = *(const v4f*)(qrow);
  v4f qv1 = *(const v4f*)(qrow + 4);
  v4f wv0 = *(const v4f*)(w_v + hbase);
  v4f wv1 = *(const v4f*)(w_v + hbase + 4);

  int vl = valid_lens[b];
  vl = vl < 1 ? 1 : (vl > KV_ ? KV_ : vl);

  const float* kbase = kp     + (size_t)b * KV_ * H_;
  const float* vbase = values + (size_t)b * KV_ * D_;

  float mrun = -3.0e38f;
  float lrun = 0.0f;
  float acc[8];
#pragma unroll
  for (int j = 0; j < 8; ++j) acc[j] = 0.0f;

  for (int kv = wid; kv < vl; kv += 4) {
    const float* krow = kbase + (size_t)kv * H_ + hbase;
    v4f k0 = *(const v4f*)(krow);
    v4f k1 = *(const v4f*)(krow + 4);

    float s = 0.0f;
#pragma unroll
    for (int j = 0; j < 4; ++j) s += wv0[j] * fast_tanh(qv0[j] + k0[j]);
#pragma unroll
    for (int j = 0; j < 4; ++j) s += wv1[j] * fast_tanh(qv1[j] + k1[j]);
#pragma unroll
    for (int off = 16; off > 0; off >>= 1) s += __shfl_xor(s, off, 32);

    const float mnew  = fmaxf(mrun, s);
    const float scale = __expf(mrun - mnew);   // V_EXP_F32
    const float p     = __expf(s - mnew);
    lrun = lrun * scale + p;

    const float* vrow = vbase + (size_t)kv * D_ + hbase;
    v4f vv0 = *(const v4f*)(vrow);
    v4f vv1 = *(const v4f*)(vrow + 4);
#pragma unroll
    for (int j = 0; j < 4; ++j) {
      acc[j]     = acc[j]     * scale + p * vv0[j];
      acc[4 + j] = acc[4 + j] * scale + p * vv1[j];
    }
    mrun = mnew;
  }

  // publish per-wave partial softmax state
#pragma unroll
  for (int j = 0; j < 8; ++j) sm_acc[wid][hbase + j] = acc[j];
  if (lane == 0) { sm_m[wid] = mrun; sm_l[wid] = lrun; }
  __syncthreads();

  // merge 4 partial (m, l, acc) triples; every thread does the cheap scalars
  const float M  = fmaxf(fmaxf(sm_m[0], sm_m[1]), fmaxf(sm_m[2], sm_m[3]));
  const float e0 = __expf(sm_m[0] - M);
  const float e1 = __expf(sm_m[1] - M);
  const float e2 = __expf(sm_m[2] - M);
  const float e3 = __expf(sm_m[3] - M);
  const float L  = e0 * sm_l[0] + e1 * sm_l[1] + e2 * sm_l[2] + e3 * sm_l[3];
  const float inv = 1.0f / L;

  float* orow = out + (size_t)bq * D_;
  for (int h = threadIdx.x; h < D_; h += 128) {
    const float s = e0 * sm_acc[0][h] + e1 * sm_acc[1][h] +
                    e2 * sm_acc[2][h] + e3 * sm_acc[3][h];
    orow[h] = s * inv;
  }
}

extern "C" void kernel_launch(void* const* d_in, const int* in_sizes, int n_in,
                              void* d_out, int out_size, void* d_ws, size_t ws_size,
                              hipStream_t stream) {
  const float* queries    = (const float*)d_in[0];  // (B,Q,D)
  const float* keys       = (const float*)d_in[1];  // (B,KV,D)
  const float* values     = (const float*)d_in[2];  // (B,KV,D)
  const int*   valid_lens = (const int*)  d_in[3];  // (B,)
  const float* W_q        = (const float*)d_in[4];  // (D,H)
  const float* W_k        = (const float*)d_in[5];  // (D,H)
  const float* w_v        = (const float*)d_in[6];  // (H,)
  float* out = (float*)d_out;

  float* qp = (float*)d_ws;              // (B*Q, H)  = 512*256 f32
  float* kp = qp + (size_t)B_ * Q_ * H_; // (B*KV, H) = 4096*256 f32

  proj_gemm_wmma<<<dim3((B_ * Q_) / 16,  H_ / 64), 32, 0, stream>>>(queries, W_q, qp);
  proj_gemm_wmma<<<dim3((B_ * KV_) / 16, H_ / 64), 32, 0, stream>>>(keys,    W_k, kp);
  attn_fused<<<B_ * Q_, 128, 0, stream>>>(qp, kp, values, valid_lens, w_v, out);
}